// RealSHT_64561948394276
// MI455X (gfx1250) — compile-verified
//
#include <hip/hip_runtime.h>

typedef __attribute__((ext_vector_type(2))) float v2f;
typedef __attribute__((ext_vector_type(8))) float v8f;

#define NLAT 361
#define NLON 720
#define LMAX 360
#define MMAX 361
#define NCH  8
#define MPAD 368            // MMAX padded to multiple of 16
#define KPAD 364            // NLAT padded to multiple of 4
#define ROWS (NCH * NLAT)   // 2888 (c,lat) rows for the DFT GEMM
#define ROWTILES 181        // ceil(2888/16)
#define LTILES 23           // ceil(360/16) (and ceil(361/16) for m)

// ---------------------------------------------------------------------------
// Stage 0: twiddle tables, laid out [m][n] so the GEMM B-operand is
// K(=n)-contiguous.  Phase reduced exactly via (m*n) mod 720 before sincos.
// Scale 2*pi/720 from the reference folded in here.
// ---------------------------------------------------------------------------
__global__ void twiddle_gen(float* __restrict__ Tcos, float* __restrict__ Tsin) {
    int idx = blockIdx.x * blockDim.x + threadIdx.x;
    if (idx >= MPAD * NLON) return;
    int m = idx / NLON;
    int n = idx - m * NLON;
    int ph = (m * n) % NLON;                       // exact in int32
    float th = (6.2831853071795864769f / (float)NLON) * (float)ph;
    float s, c;
    sincosf(th, &s, &c);
    const float scale = 6.2831853071795864769f / (float)NLON;
    Tcos[idx] =  c * scale;                        // Re(exp(-i th)) = cos
    Tsin[idx] = -s * scale;                        // Im(exp(-i th)) = -sin
}

// ---------------------------------------------------------------------------
// Stage 1: rfft as NT-GEMM with V_WMMA_F32_16X16X4_F32.
//   A: x rows (c,lat), K = n (720, multiple of 4)
//   B: Tcos/Tsin rows m, K = n
// Two accumulators (real/imag) share every A load.
// Output scattered into Xbuf[m][r][k] (r: 0-7 real ch, 8-15 imag ch),
// which is exactly the A layout stage 2 consumes.
// ---------------------------------------------------------------------------
__global__ __launch_bounds__(32) void dft_wmma(
        const float* __restrict__ x,
        const float* __restrict__ Tcos,
        const float* __restrict__ Tsin,
        float* __restrict__ Xbuf) {
    const int mt   = blockIdx.x;        // row tile 0..180
    const int nt   = blockIdx.y;        // m  tile 0..22
    const int lane = threadIdx.x;       // wave32
    const int half = lane >> 4;         // 0/1
    const int ko   = half << 1;         // K sub-offset 0/2
    const int lr   = lane & 15;

    int arow  = mt * 16 + lr;
    int arowc = arow < ROWS ? arow : ROWS - 1;          // clamp pad rows (stores guarded)
    const float* ap = x    + (size_t)arowc * NLON + ko;
    const int bcol  = nt * 16 + lr;                     // < MPAD by construction
    const float* cp = Tcos + (size_t)bcol * NLON + ko;
    const float* sp = Tsin + (size_t)bcol * NLON + ko;

    v8f accR = {0.f,0.f,0.f,0.f,0.f,0.f,0.f,0.f};
    v8f accI = {0.f,0.f,0.f,0.f,0.f,0.f,0.f,0.f};
    for (int kk = 0; kk < NLON; kk += 4) {
        v2f a  = *(const v2f*)(ap + kk);
        v2f bc = *(const v2f*)(cp + kk);
        v2f bs = *(const v2f*)(sp + kk);
        accR = __builtin_amdgcn_wmma_f32_16x16x4_f32(false, a, false, bc,
                                                     (short)0, accR, false, false);
        accI = __builtin_amdgcn_wmma_f32_16x16x4_f32(false, a, false, bs,
                                                     (short)0, accI, false, false);
    }

    const int m = nt * 16 + lr;                         // C/D column = lane&15
    if (m < MMAX) {
        #pragma unroll
        for (int j = 0; j < 8; ++j) {
            int row = mt * 16 + j + 8 * half;           // C/D row = j + 8*(lane>>4)
            if (row < ROWS) {
                int c = row / NLAT;
                int k = row - c * NLAT;
                Xbuf[((size_t)m * 16 + c    ) * KPAD + k] = accR[j];
                Xbuf[((size_t)m * 16 + c + 8) * KPAD + k] = accI[j];
            }
        }
    }
}

// ---------------------------------------------------------------------------
// Stage 2: per-m contraction out[c,l,m] = sum_k X[c,k,m] * W[m,l,k].
// One wave per (m, 16-wide l tile).  B = weights[m][l][:] is already
// [col][K]-contiguous in memory, streamed once from HBM (bandwidth-bound).
// K=361 = 90 full WMMA steps + one zero-padded tail step (no OOB reads).
// ---------------------------------------------------------------------------
__global__ __launch_bounds__(32) void einsum_wmma(
        const float* __restrict__ W,
        const float* __restrict__ Xbuf,
        float* __restrict__ out) {
    const int m    = blockIdx.x;        // 0..360
    const int lt   = blockIdx.y;        // 0..22
    const int lane = threadIdx.x;
    const int half = lane >> 4;
    const int ko   = half << 1;
    const int lr   = lane & 15;

    const float* A  = Xbuf + (size_t)m * 16 * KPAD;
    const float* ap = A + (size_t)lr * KPAD + ko;
    const int l  = lt * 16 + lr;
    const int lc = l < LMAX ? l : LMAX - 1;             // clamp pad cols (stores guarded)
    const float* bp = W + ((size_t)m * LMAX + lc) * NLAT + ko;

    v8f acc = {0.f,0.f,0.f,0.f,0.f,0.f,0.f,0.f};
    for (int kk = 0; kk < 360; kk += 4) {               // k = 0..359
        v2f a = *(const v2f*)(ap + kk);
        v2f b = *(const v2f*)(bp + kk);
        acc = __builtin_amdgcn_wmma_f32_16x16x4_f32(false, a, false, b,
                                                    (short)0, acc, false, false);
    }
    // Tail: k = 360 only (valid in lane-half 0, element 0); others contribute 0.
    {
        v2f a = {0.f, 0.f};
        v2f b = {0.f, 0.f};
        if (half == 0) {
            a[0] = A[(size_t)lr * KPAD + 360];
            b[0] = W[((size_t)m * LMAX + lc) * NLAT + 360];
        }
        acc = __builtin_amdgcn_wmma_f32_16x16x4_f32(false, a, false, b,
                                                    (short)0, acc, false, false);
    }

    if (l < LMAX) {
        #pragma unroll
        for (int j = 0; j < 8; ++j) {
            int r  = j + 8 * half;                      // 0-7: real, 8-15: imag
            int c  = r & 7;
            int im = r >> 3;
            out[2 * (((size_t)c * LMAX + l) * MMAX + m) + im] = acc[j];
        }
    }
}

// ---------------------------------------------------------------------------
extern "C" void kernel_launch(void* const* d_in, const int* in_sizes, int n_in,
                              void* d_out, int out_size, void* d_ws, size_t ws_size,
                              hipStream_t stream) {
    const float* x = (const float*)d_in[0];   // (1, 8, 361, 720) f32
    const float* w = (const float*)d_in[1];   // (361, 360, 361)  f32
    float* out = (float*)d_out;               // (1, 8, 360, 361) complex64 interleaved

    float* Tcos = (float*)d_ws;               // MPAD*NLON
    float* Tsin = Tcos + (size_t)MPAD * NLON; // MPAD*NLON
    float* Xbuf = Tsin + (size_t)MPAD * NLON; // MMAX*16*KPAD  (~8.4 MB)

    const int ntw = MPAD * NLON;
    twiddle_gen<<<(ntw + 255) / 256, 256, 0, stream>>>(Tcos, Tsin);

    dim3 g1(ROWTILES, LTILES);                // 181 x 23 (row tiles x m tiles)
    dft_wmma<<<g1, 32, 0, stream>>>(x, Tcos, Tsin, Xbuf);

    dim3 g2(MMAX, LTILES);                    // 361 x 23 (m x l tiles)
    einsum_wmma<<<g2, 32, 0, stream>>>(w, Xbuf, out);
}